// targetPredict_1717986918487
// MI455X (gfx1250) — compile-verified
//
#include <hip/hip_runtime.h>
#include <math.h>

#define N_OBJ 21
#define N_ACT 27
#define HUMAN 14
#define RCOUNT 2000
#define HF 38
#define WF 50
#define D_IN 25088   // 512*7*7
#define D_HID 4096
#define NEGV (-1e30f)
#define KCH 4        // K-chunks (waves) per output tile

// workspace layout (float offsets)
#define OFF_M    0        // 2000: per-row max of scores
#define OFF_SE   2000     // 2000: per-row sum(exp(s-m))
#define OFF_BOX  4000     // 20*4: per-class winner boxes (feature coords, yxyx)
#define OFF_POOL 4096     // 25088: pooled feature vector
#define OFF_H1   29184    // 4096
#define OFF_FC7  33280    // 4096
#define OFF_LOC  37376    // 4: pred_object_loc

typedef float v2f __attribute__((ext_vector_type(2)));
typedef float v8f __attribute__((ext_vector_type(8)));

__device__ __forceinline__ int clampi(int v, int lo, int hi) {
    return v < lo ? lo : (v > hi ? hi : v);
}

// ---------------- K1: per-row softmax statistics ----------------
__global__ void row_stats_kernel(const float* __restrict__ s, float* __restrict__ ws) {
    int r = blockIdx.x * blockDim.x + threadIdx.x;
    if (r >= RCOUNT) return;
    float m = -INFINITY;
    #pragma unroll
    for (int c = 0; c < N_OBJ; ++c) m = fmaxf(m, s[r * N_OBJ + c]);
    float se = 0.f;
    #pragma unroll
    for (int c = 0; c < N_OBJ; ++c) se += expf(s[r * N_OBJ + c] - m);
    ws[OFF_M + r] = m;
    ws[OFF_SE + r] = se;
}

// ---------------- K2: per-class argmax of prob + box decode ----------------
// NMS with thresh 0.0 keeps exactly the top-prob box per class.
__global__ void class_pick_kernel(const float* __restrict__ s,
                                  const float* __restrict__ off,
                                  const float* __restrict__ rois,
                                  const int* __restrict__ img,
                                  float* __restrict__ ws) {
    __shared__ float sp[256];
    __shared__ int   si[256];
    int lab = blockIdx.x;      // 0..19
    int c = lab + 1;           // class column 1..20
    int tid = threadIdx.x;
    float bp = -1.f; int br = 0x7fffffff;
    for (int r = tid; r < RCOUNT; r += 256) {
        float p = expf(s[r * N_OBJ + c] - ws[OFF_M + r]) / ws[OFF_SE + r];
        if (p > bp || (p == bp && r < br)) { bp = p; br = r; }
    }
    sp[tid] = bp; si[tid] = br;
    __syncthreads();
    for (int st = 128; st > 0; st >>= 1) {
        if (tid < st) {
            float p2 = sp[tid + st]; int r2 = si[tid + st];
            if (p2 > sp[tid] || (p2 == sp[tid] && r2 < si[tid])) { sp[tid] = p2; si[tid] = r2; }
        }
        __syncthreads();
    }
    if (tid == 0) {
        int r = si[0];
        float sy = (float)HF / (float)img[0];
        float sx = (float)WF / (float)img[1];
        float y1 = rois[r * 4 + 0] * sy, x1 = rois[r * 4 + 1] * sx;
        float y2 = rois[r * 4 + 2] * sy, x2 = rois[r * 4 + 3] * sx;
        float h = y2 - y1, w = x2 - x1;
        float cy = y1 + 0.5f * h, cx = x1 + 0.5f * w;
        const float* o = off + (size_t)r * (N_OBJ * 4) + c * 4;
        float dy = o[0] * 0.1f, dx = o[1] * 0.1f, dh = o[2] * 0.2f, dw = o[3] * 0.2f;
        float cy2 = dy * h + cy, cx2 = dx * w + cx;
        float h2 = expf(dh) * h, w2 = expf(dw) * w;
        ws[OFF_BOX + lab * 4 + 0] = fminf(fmaxf(cy2 - 0.5f * h2, 0.f), (float)HF);
        ws[OFF_BOX + lab * 4 + 1] = fminf(fmaxf(cx2 - 0.5f * w2, 0.f), (float)WF);
        ws[OFF_BOX + lab * 4 + 2] = fminf(fmaxf(cy2 + 0.5f * h2, 0.f), (float)HF);
        ws[OFF_BOX + lab * 4 + 3] = fminf(fmaxf(cx2 + 0.5f * w2, 0.f), (float)WF);
    }
}

// ---------------- K3: RoI max pool of the human box ----------------
__global__ void roipool_kernel(const float* __restrict__ x, float* __restrict__ ws) {
    int idx = blockIdx.x * blockDim.x + threadIdx.x;
    if (idx >= D_IN) return;
    int c = idx / 49, rem = idx % 49, ph = rem / 7, pw = rem % 7;
    const float* hb = ws + OFF_BOX + HUMAN * 4;
    float y1 = hb[0], x1 = hb[1], y2 = hb[2], x2 = hb[3];
    // jnp.round == round-half-even -> rintf
    int xmin = (int)rintf(x1 * 0.0625f), ymin = (int)rintf(y1 * 0.0625f);
    int xmax = (int)rintf(x2 * 0.0625f), ymax = (int)rintf(y2 * 0.0625f);
    float rw = (float)((xmax - xmin + 1) > 1 ? (xmax - xmin + 1) : 1);
    float rh = (float)((ymax - ymin + 1) > 1 ? (ymax - ymin + 1) : 1);
    int hs = clampi(ymin + (int)floorf((float)ph * rh / 7.f), 0, HF);
    int he = clampi(ymin + (int)ceilf((float)(ph + 1) * rh / 7.f), 0, HF);
    int wsrt = clampi(xmin + (int)floorf((float)pw * rw / 7.f), 0, WF);
    int wend = clampi(xmin + (int)ceilf((float)(pw + 1) * rw / 7.f), 0, WF);
    float m = NEGV;
    for (int hh = hs; hh < he; ++hh)
        for (int wi = wsrt; wi < wend; ++wi)
            m = fmaxf(m, x[(size_t)c * (HF * WF) + hh * WF + wi]);
    ws[OFF_POOL + idx] = (m <= NEGV * 0.5f) ? 0.f : m;
}

// ---------------- K4/K5: fp32 WMMA GEMV (M padded 1->16), out = relu(v@W + b) ----------------
// 256 blocks x 4 waves. Each wave accumulates one quarter of K for a 16-output
// tile with V_WMMA_F32_16X16X4_F32; LDS reduction combines the 4 partials.
// A 16x4 f32 layout: lanes 0-15 hold (M=lane, K=0/1) in the 2 VGPRs, lanes
// 16-31 hold K=2/3. Only M=0 is nonzero -> uniform scalar loads of the input
// vector, selected into lanes 0/16 branch-free (EXEC stays all-ones).
// B 4x16 f32: lanes 0-15 = N for K=0(.x)/1(.y); lanes 16-31 for K=2/3.
// D row M=0 lands in acc[0] of lanes 0-15.
__global__ void wmma_gemv_kernel(const float* __restrict__ W,
                                 const float* __restrict__ vin,
                                 const float* __restrict__ bias,
                                 float* __restrict__ vout, int K) {
    __shared__ float red[KCH][16];
    int lane = threadIdx.x & 31;
    int wave = threadIdx.x >> 5;
    int nb = blockIdx.x * 16;
    int n = nb + (lane & 15);
    int kk = (lane < 16) ? 0 : 2;
    int Kq = K / KCH;                       // 6272 or 1024, multiple of 16
    int kstart = wave * Kq;

    const float* Wp = W + (size_t)(kstart + kk) * D_HID + n;  // per-lane base
    const float* vp = vin + kstart;                           // uniform base
    bool lo = (lane < 16);
    float msk = ((lane & 15) == 0) ? 1.f : 0.f;               // lanes 0 and 16

    v8f acc = {};
    int iters = Kq >> 4;                    // 16 K-values per iteration
    for (int it = 0; it < iters; ++it) {
        #pragma unroll
        for (int u = 0; u < 4; ++u) {
            int kb = u * 4;
            // uniform scalar loads -> s_load; branch-free lane placement
            float a0 = lo ? vp[kb]     : vp[kb + 2];
            float a1 = lo ? vp[kb + 1] : vp[kb + 3];
            v2f a; a.x = a0 * msk; a.y = a1 * msk;
            v2f b;
            b.x = Wp[(size_t)kb * D_HID];            // imm offsets: u*64KB
            b.y = Wp[(size_t)kb * D_HID + D_HID];    // and +16KB
            acc = __builtin_amdgcn_wmma_f32_16x16x4_f32(
                false, a, false, b, (short)0, acc, false, false);
        }
        Wp += (size_t)16 * D_HID;
        vp += 16;
    }
    if (lane < 16) red[wave][lane] = acc[0];
    __syncthreads();
    if (threadIdx.x < 16) {
        int t = threadIdx.x;
        float v = red[0][t] + red[1][t] + red[2][t] + red[3][t] + bias[nb + t];
        vout[nb + t] = fmaxf(v, 0.f);
    }
}

// ---------------- K6: heads (loc: 4 cols, act: 27 cols) ----------------
__global__ void heads_kernel(const float* __restrict__ fc7,
                             const float* __restrict__ Wl, const float* __restrict__ bl,
                             const float* __restrict__ Wa, const float* __restrict__ ba,
                             float* __restrict__ ws, float* __restrict__ out) {
    __shared__ float sm[256];
    int j = blockIdx.x;        // 0..30
    int tid = threadIdx.x;
    float p = 0.f;
    if (j < 4) {
        for (int k = tid; k < D_HID; k += 256) p += fc7[k] * Wl[(size_t)k * 4 + j];
    } else {
        int ja = j - 4;
        for (int k = tid; k < D_HID; k += 256) p += fc7[k] * Wa[(size_t)k * N_ACT + ja];
    }
    sm[tid] = p; __syncthreads();
    for (int st = 128; st > 0; st >>= 1) {
        if (tid < st) sm[tid] += sm[tid + st];
        __syncthreads();
    }
    if (tid == 0) {
        if (j < 4) ws[OFF_LOC + j] = sm[0] + bl[j];
        else       out[8 + (j - 4)] = sm[0] + ba[j - 4];
    }
}

// ---------------- K7: gaussian pick + output assembly ----------------
__global__ void finalize_kernel(const int* __restrict__ img,
                                const float* __restrict__ ws, float* __restrict__ out) {
    if (threadIdx.x != 0 || blockIdx.x != 0) return;
    float invy = (float)img[0] / (float)HF;
    float invx = (float)img[1] / (float)WF;
    const float* hb = ws + OFF_BOX + HUMAN * 4;
    out[0] = hb[0] * invy; out[1] = hb[1] * invx;
    out[2] = hb[2] * invy; out[3] = hb[3] * invx;
    // xywh quirk: [0.5w, 0.5h, w, h] with w=b3-b1, h=b2-b0
    float hw0 = hb[3] - hb[1], hh0 = hb[2] - hb[0];
    float h40 = 0.5f * hw0, h41 = 0.5f * hh0;
    float hwm = fmaxf(hw0, 1e-3f), hhm = fmaxf(hh0, 1e-3f);
    const float* pl = ws + OFF_LOC;
    float mw = pl[3] - pl[1], mh = pl[2] - pl[0];
    float mu0 = 0.5f * mw, mu1 = 0.5f * mh, mu2 = mw, mu3 = mh;
    float bg = -1.f; int bidx = 0;
    for (int lab = 0; lab < 20; ++lab) {
        if (lab == HUMAN) continue;
        const float* bb = ws + OFF_BOX + lab * 4;
        float cw0 = bb[3] - bb[1], ch0 = bb[2] - bb[0];
        float cwm = fmaxf(cw0, 1e-3f), chm = fmaxf(ch0, 1e-3f);
        float b0 = (0.5f * cw0 - h40) / hwm;
        float b1 = (0.5f * ch0 - h41) / hhm;
        float b2 = logf(cwm / hwm);
        float b3 = logf(chm / hhm);
        float d0 = b0 - mu0, d1 = b1 - mu1, d2 = b2 - mu2, d3 = b3 - mu3;
        float g = expf(-(d0 * d0 + d1 * d1 + d2 * d2 + d3 * d3) / (2.f * 0.3f * 0.3f));
        if (g > bg) { bg = g; bidx = lab; }   // strict > == first-index tie-break
    }
    const float* ob = ws + OFF_BOX + bidx * 4;
    out[4] = ob[0] * invy; out[5] = ob[1] * invx;
    out[6] = ob[2] * invy; out[7] = ob[3] * invx;
    out[35] = (float)HF / (float)img[0];      // my_scale
}

extern "C" void kernel_launch(void* const* d_in, const int* in_sizes, int n_in,
                              void* d_out, int out_size, void* d_ws, size_t ws_size,
                              hipStream_t stream) {
    const float* x      = (const float*)d_in[0];
    const float* scores = (const float*)d_in[1];
    const float* off    = (const float*)d_in[2];
    const float* rois   = (const float*)d_in[3];
    const int*   img    = (const int*)d_in[4];
    const float* W1     = (const float*)d_in[5];
    const float* b1     = (const float*)d_in[6];
    const float* W2     = (const float*)d_in[7];
    const float* b2     = (const float*)d_in[8];
    const float* Wloc   = (const float*)d_in[9];
    const float* bloc   = (const float*)d_in[10];
    const float* Wact   = (const float*)d_in[11];
    const float* bact   = (const float*)d_in[12];
    float* out = (float*)d_out;
    float* ws  = (float*)d_ws;

    row_stats_kernel<<<(RCOUNT + 255) / 256, 256, 0, stream>>>(scores, ws);
    class_pick_kernel<<<20, 256, 0, stream>>>(scores, off, rois, img, ws);
    roipool_kernel<<<(D_IN + 255) / 256, 256, 0, stream>>>(x, ws);
    // layer 1: 25088 -> 4096 (411 MB of W1 streamed once; bandwidth-bound)
    wmma_gemv_kernel<<<D_HID / 16, 32 * KCH, 0, stream>>>(W1, ws + OFF_POOL, b1, ws + OFF_H1, D_IN);
    // layer 2: 4096 -> 4096
    wmma_gemv_kernel<<<D_HID / 16, 32 * KCH, 0, stream>>>(W2, ws + OFF_H1, b2, ws + OFF_FC7, D_HID);
    heads_kernel<<<4 + N_ACT, 256, 0, stream>>>(ws + OFF_FC7, Wloc, bloc, Wact, bact, ws, out);
    finalize_kernel<<<1, 32, 0, stream>>>(img, ws, out);
}